// MPMDiffSim_85194971283653
// MI455X (gfx1250) — compile-verified
//
#include <hip/hip_runtime.h>

// ---------------- constants (match reference exactly) ----------------
#define NG        64
#define GSZ       (NG * NG * NG)
#define NTHREADS  256
#define BOUNDARY  3

__device__ constexpr float DXf    = 1.0f / 64.0f;
__device__ constexpr float INVDX  = 64.0f;
__device__ constexpr float DTc    = 5e-4f;
__device__ constexpr float PVOL   = (1.0f/128.0f) * (1.0f/128.0f) * (1.0f/128.0f);
__device__ constexpr float PMASS  = PVOL * 1000.0f;
__device__ constexpr float AFFC   = -DTc * PVOL * 4.0f * INVDX * INVDX;   // stress coefficient
__device__ constexpr float CSCALE = 4.0f * INVDX * INVDX;                 // C_next scale

// ---------------- CDNA5 async global->LDS copy helpers ----------------
typedef __attribute__((address_space(1))) int g_int;
typedef __attribute__((address_space(3))) int l_int;

__device__ __forceinline__ void async_copy_f32(const float* gsrc, float* lds_dst) {
#if __has_builtin(__builtin_amdgcn_global_load_async_to_lds_b32)
    __builtin_amdgcn_global_load_async_to_lds_b32(
        (g_int*)gsrc, (l_int*)lds_dst, 0, 0);
#else
    // VDST = LDS byte address (low 32 bits of generic shared pointer),
    // VADDR = 64-bit global address.
    unsigned int laddr = (unsigned int)(unsigned long long)lds_dst;
    asm volatile("global_load_async_to_lds_b32 %0, %1, off"
                 :: "v"(laddr), "v"(gsrc) : "memory");
#endif
}

__device__ __forceinline__ void async_wait_all() {
#if __has_builtin(__builtin_amdgcn_s_wait_asynccnt)
    __builtin_amdgcn_s_wait_asynccnt(0);
#else
    asm volatile("s_wait_asynccnt 0" ::: "memory");
#endif
}

// ---------------- kernel: P2G scatter (first in file so disasm snippet shows it) ----------------
__global__ __launch_bounds__(NTHREADS) void mpm_p2g(
    const float* __restrict__ x, const float* __restrict__ v,
    const float* __restrict__ C, const float* __restrict__ stress,
    float* __restrict__ grid_mv, float* __restrict__ grid_m, int n)
{
    __shared__ float s_x[NTHREADS * 3];
    __shared__ float s_v[NTHREADS * 3];
    __shared__ float s_C[NTHREADS * 9];
    __shared__ float s_s[NTHREADS * 9];

    const int tid = threadIdx.x;
    const int blk = blockIdx.x * NTHREADS;
    const int cnt = (n - blk < NTHREADS) ? (n - blk) : NTHREADS;

    // async-stage this block's contiguous particle records into LDS (ASYNCcnt path)
    for (int i = tid; i < cnt * 3; i += NTHREADS) async_copy_f32(x + (size_t)blk * 3 + i, &s_x[i]);
    for (int i = tid; i < cnt * 3; i += NTHREADS) async_copy_f32(v + (size_t)blk * 3 + i, &s_v[i]);
    for (int i = tid; i < cnt * 9; i += NTHREADS) async_copy_f32(C + (size_t)blk * 9 + i, &s_C[i]);
    for (int i = tid; i < cnt * 9; i += NTHREADS) async_copy_f32(stress + (size_t)blk * 9 + i, &s_s[i]);
    async_wait_all();
    __syncthreads();

    if (tid < cnt) {
        const float x0 = s_x[tid*3+0], x1 = s_x[tid*3+1], x2 = s_x[tid*3+2];
        const int   b0 = (int)floorf(x0 * INVDX - 0.5f);
        const int   b1 = (int)floorf(x1 * INVDX - 0.5f);
        const int   b2 = (int)floorf(x2 * INVDX - 0.5f);
        const float f0 = x0 * INVDX - (float)b0;
        const float f1 = x1 * INVDX - (float)b1;
        const float f2 = x2 * INVDX - (float)b2;

        float wx[3], wy[3], wz[3], dpx[3], dpy[3], dpz[3];
        wx[0]=0.5f*(1.5f-f0)*(1.5f-f0); wx[1]=0.75f-(f0-1.f)*(f0-1.f); wx[2]=0.5f*(f0-0.5f)*(f0-0.5f);
        wy[0]=0.5f*(1.5f-f1)*(1.5f-f1); wy[1]=0.75f-(f1-1.f)*(f1-1.f); wy[2]=0.5f*(f1-0.5f)*(f1-0.5f);
        wz[0]=0.5f*(1.5f-f2)*(1.5f-f2); wz[1]=0.75f-(f2-1.f)*(f2-1.f); wz[2]=0.5f*(f2-0.5f)*(f2-0.5f);
        #pragma unroll
        for (int o = 0; o < 3; ++o) {
            dpx[o] = ((float)o - f0) * DXf;
            dpy[o] = ((float)o - f1) * DXf;
            dpz[o] = ((float)o - f2) * DXf;
        }

        float aff[9];
        #pragma unroll
        for (int i = 0; i < 9; ++i)
            aff[i] = s_s[tid*9+i] * AFFC + PMASS * s_C[tid*9+i];

        const float pmv0 = PMASS * s_v[tid*3+0];
        const float pmv1 = PMASS * s_v[tid*3+1];
        const float pmv2 = PMASS * s_v[tid*3+2];

        #pragma unroll
        for (int i = 0; i < 3; ++i) {
            const float wi  = wx[i], d0 = dpx[i];
            const int   n0  = (b0 + i) * NG;
            #pragma unroll
            for (int j = 0; j < 3; ++j) {
                const float wij = wi * wy[j], d1 = dpy[j];
                const int   n01 = (n0 + b1 + j) * NG + b2;
                const float m0p = pmv0 + aff[0]*d0 + aff[1]*d1;
                const float m1p = pmv1 + aff[3]*d0 + aff[4]*d1;
                const float m2p = pmv2 + aff[6]*d0 + aff[7]*d1;
                #pragma unroll
                for (int k = 0; k < 3; ++k) {
                    const float w3 = wij * wz[k], d2 = dpz[k];
                    const int node = n01 + k;
                    unsafeAtomicAdd(grid_mv + (size_t)node*3 + 0, w3 * (m0p + aff[2]*d2));
                    unsafeAtomicAdd(grid_mv + (size_t)node*3 + 1, w3 * (m1p + aff[5]*d2));
                    unsafeAtomicAdd(grid_mv + (size_t)node*3 + 2, w3 * (m2p + aff[8]*d2));
                    unsafeAtomicAdd(grid_m + node, w3 * PMASS);
                }
            }
        }
    }
}

// ---------------- kernel: zero the grid (grid_mv[3G] + grid_m[G]) ----------------
__global__ __launch_bounds__(NTHREADS) void mpm_zero_grid(float4* __restrict__ g) {
    int i = blockIdx.x * NTHREADS + threadIdx.x;   // GSZ float4s total (4*GSZ floats)
    if (i < GSZ) g[i] = float4{0.f, 0.f, 0.f, 0.f};
}

// ---------------- kernel: grid normalize + gravity + boundary ----------------
__global__ __launch_bounds__(NTHREADS) void mpm_grid_update(
    float* __restrict__ grid_mv, const float* __restrict__ grid_m)
{
    int g = blockIdx.x * NTHREADS + threadIdx.x;
    if (g >= GSZ) return;
    const float m = grid_m[g];
    float v0 = 0.f, v1 = 0.f, v2 = 0.f;
    if (m > 0.f) {
        const float inv = 1.f / fmaxf(m, 1e-10f);
        v0 = grid_mv[(size_t)g*3 + 0] * inv;
        v1 = grid_mv[(size_t)g*3 + 1] * inv + DTc * (-9.8f);
        v2 = grid_mv[(size_t)g*3 + 2] * inv;
        const int gi = g >> 12, gj = (g >> 6) & (NG - 1), gk = g & (NG - 1);
        if ((gi < BOUNDARY && v0 < 0.f) || (gi >= NG - BOUNDARY && v0 > 0.f)) v0 = 0.f;
        if ((gj < BOUNDARY && v1 < 0.f) || (gj >= NG - BOUNDARY && v1 > 0.f)) v1 = 0.f;
        if ((gk < BOUNDARY && v2 < 0.f) || (gk >= NG - BOUNDARY && v2 > 0.f)) v2 = 0.f;
    }
    grid_mv[(size_t)g*3 + 0] = v0;   // grid_v stored in place
    grid_mv[(size_t)g*3 + 1] = v1;
    grid_mv[(size_t)g*3 + 2] = v2;
}

// ---------------- kernel: G2P gather + advect + F update ----------------
__global__ __launch_bounds__(NTHREADS) void mpm_g2p(
    const float* __restrict__ x, const float* __restrict__ F,
    const float* __restrict__ grid_v,
    float* __restrict__ out_x, float* __restrict__ out_v,
    float* __restrict__ out_C, float* __restrict__ out_F, int n)
{
    __shared__ float s_x[NTHREADS * 3];
    __shared__ float s_F[NTHREADS * 9];

    const int tid = threadIdx.x;
    const int blk = blockIdx.x * NTHREADS;
    const int cnt = (n - blk < NTHREADS) ? (n - blk) : NTHREADS;

    for (int i = tid; i < cnt * 3; i += NTHREADS) async_copy_f32(x + (size_t)blk * 3 + i, &s_x[i]);
    for (int i = tid; i < cnt * 9; i += NTHREADS) async_copy_f32(F + (size_t)blk * 9 + i, &s_F[i]);
    async_wait_all();
    __syncthreads();

    if (tid < cnt) {
        const int   p  = blk + tid;
        const float x0 = s_x[tid*3+0], x1 = s_x[tid*3+1], x2 = s_x[tid*3+2];
        const int   b0 = (int)floorf(x0 * INVDX - 0.5f);
        const int   b1 = (int)floorf(x1 * INVDX - 0.5f);
        const int   b2 = (int)floorf(x2 * INVDX - 0.5f);
        const float f0 = x0 * INVDX - (float)b0;
        const float f1 = x1 * INVDX - (float)b1;
        const float f2 = x2 * INVDX - (float)b2;

        float wx[3], wy[3], wz[3], dpx[3], dpy[3], dpz[3];
        wx[0]=0.5f*(1.5f-f0)*(1.5f-f0); wx[1]=0.75f-(f0-1.f)*(f0-1.f); wx[2]=0.5f*(f0-0.5f)*(f0-0.5f);
        wy[0]=0.5f*(1.5f-f1)*(1.5f-f1); wy[1]=0.75f-(f1-1.f)*(f1-1.f); wy[2]=0.5f*(f1-0.5f)*(f1-0.5f);
        wz[0]=0.5f*(1.5f-f2)*(1.5f-f2); wz[1]=0.75f-(f2-1.f)*(f2-1.f); wz[2]=0.5f*(f2-0.5f)*(f2-0.5f);
        #pragma unroll
        for (int o = 0; o < 3; ++o) {
            dpx[o] = ((float)o - f0) * DXf;
            dpy[o] = ((float)o - f1) * DXf;
            dpz[o] = ((float)o - f2) * DXf;
        }

        float nv0=0.f, nv1=0.f, nv2=0.f;
        float c00=0.f,c01=0.f,c02=0.f,c10=0.f,c11=0.f,c12=0.f,c20=0.f,c21=0.f,c22=0.f;

        #pragma unroll
        for (int i = 0; i < 3; ++i) {
            const float wi = wx[i], d0 = dpx[i];
            const int   n0 = (b0 + i) * NG;
            #pragma unroll
            for (int j = 0; j < 3; ++j) {
                const float wij = wi * wy[j], d1 = dpy[j];
                const int   n01 = (n0 + b1 + j) * NG + b2;
                #pragma unroll
                for (int k = 0; k < 3; ++k) {
                    const float w3 = wij * wz[k], d2 = dpz[k];
                    const size_t node = (size_t)(n01 + k);
                    const float g0 = grid_v[node*3+0];
                    const float g1 = grid_v[node*3+1];
                    const float g2 = grid_v[node*3+2];
                    nv0 += w3 * g0; nv1 += w3 * g1; nv2 += w3 * g2;
                    const float wg0 = w3*g0, wg1 = w3*g1, wg2 = w3*g2;
                    c00 += wg0*d0; c01 += wg0*d1; c02 += wg0*d2;
                    c10 += wg1*d0; c11 += wg1*d1; c12 += wg1*d2;
                    c20 += wg2*d0; c21 += wg2*d1; c22 += wg2*d2;
                }
            }
        }

        c00*=CSCALE; c01*=CSCALE; c02*=CSCALE;
        c10*=CSCALE; c11*=CSCALE; c12*=CSCALE;
        c20*=CSCALE; c21*=CSCALE; c22*=CSCALE;

        out_v[(size_t)p*3+0] = nv0; out_v[(size_t)p*3+1] = nv1; out_v[(size_t)p*3+2] = nv2;
        out_x[(size_t)p*3+0] = x0 + DTc*nv0;
        out_x[(size_t)p*3+1] = x1 + DTc*nv1;
        out_x[(size_t)p*3+2] = x2 + DTc*nv2;
        out_C[(size_t)p*9+0]=c00; out_C[(size_t)p*9+1]=c01; out_C[(size_t)p*9+2]=c02;
        out_C[(size_t)p*9+3]=c10; out_C[(size_t)p*9+4]=c11; out_C[(size_t)p*9+5]=c12;
        out_C[(size_t)p*9+6]=c20; out_C[(size_t)p*9+7]=c21; out_C[(size_t)p*9+8]=c22;

        const float a00 = 1.f + DTc*c00, a01 = DTc*c01,       a02 = DTc*c02;
        const float a10 = DTc*c10,       a11 = 1.f + DTc*c11, a12 = DTc*c12;
        const float a20 = DTc*c20,       a21 = DTc*c21,       a22 = 1.f + DTc*c22;
        #pragma unroll
        for (int k = 0; k < 3; ++k) {
            const float fk0 = s_F[tid*9 + 0*3 + k];
            const float fk1 = s_F[tid*9 + 1*3 + k];
            const float fk2 = s_F[tid*9 + 2*3 + k];
            out_F[(size_t)p*9 + 0*3 + k] = a00*fk0 + a01*fk1 + a02*fk2;
            out_F[(size_t)p*9 + 1*3 + k] = a10*fk0 + a11*fk1 + a12*fk2;
            out_F[(size_t)p*9 + 2*3 + k] = a20*fk0 + a21*fk1 + a22*fk2;
        }
    }
}

// ---------------- host launcher ----------------
extern "C" void kernel_launch(void* const* d_in, const int* in_sizes, int n_in,
                              void* d_out, int out_size, void* d_ws, size_t ws_size,
                              hipStream_t stream) {
    (void)n_in; (void)out_size; (void)ws_size;
    const float* x      = (const float*)d_in[0];
    const float* v      = (const float*)d_in[1];
    const float* C      = (const float*)d_in[2];
    const float* F      = (const float*)d_in[3];
    const float* stress = (const float*)d_in[4];
    const int n = in_sizes[0] / 3;

    float* grid_mv = (float*)d_ws;            // 3*GSZ floats
    float* grid_m  = grid_mv + (size_t)3*GSZ; // GSZ floats  (total 4 MB)

    float* out   = (float*)d_out;
    float* out_x = out;
    float* out_v = out + (size_t)3*n;
    float* out_C = out + (size_t)6*n;
    float* out_F = out + (size_t)15*n;

    const int pblocks = (n + NTHREADS - 1) / NTHREADS;

    mpm_zero_grid<<<GSZ / NTHREADS, NTHREADS, 0, stream>>>((float4*)d_ws);
    mpm_p2g<<<pblocks, NTHREADS, 0, stream>>>(x, v, C, stress, grid_mv, grid_m, n);
    mpm_grid_update<<<GSZ / NTHREADS, NTHREADS, 0, stream>>>(grid_mv, grid_m);
    mpm_g2p<<<pblocks, NTHREADS, 0, stream>>>(x, F, grid_mv, out_x, out_v, out_C, out_F, n);
}